// ReconstructionModel_3118146257523
// MI455X (gfx1250) — compile-verified
//
#include <hip/hip_runtime.h>

typedef __attribute__((ext_vector_type(2))) float v2f;
typedef __attribute__((ext_vector_type(8))) float v8f;

#define NPIX 12288   // 12 * 32 * 32
#define NV   6144    // N_PIX / 2
#define ROWS 128     // 8 * 16 batch*channel rows
#define KB   64      // K chunk staged in LDS
#define COLS 32      // output columns per block
#define SA   68      // A_lds row stride in floats  (conflict-free b64 frag loads)
#define ST   68      // Bt_lds col stride in floats (conflict-free b64 frag loads)

__global__ __launch_bounds__(256)
void ellipse_mix_gemm(const float* __restrict__ X,
                      const float* __restrict__ R,
                      float* __restrict__ out)
{
    __shared__ float Alds[ROWS * SA];   // 34,816 B
    __shared__ float Btld[COLS * ST];   //  8,704 B  (B transposed: Btld[n*ST + k])

    const int tid   = threadIdx.x;
    const int lane  = tid & 31;
    const int wave  = tid >> 5;                 // 0..7 -> M tile index
    const int nbase = blockIdx.x * COLS;        // output column base

    // ---- zero the invalid tail [NV, NPIX) for this block's 32-col slice ----
    {
        const int tb = NV + nbase;
        const float4 z = {0.f, 0.f, 0.f, 0.f};
        #pragma unroll
        for (int i = 0; i < 4; ++i) {           // 128 rows x 32 cols = 1024 float4
            int f   = tid + i * 256;
            int row = f >> 3;                   // f / 8
            int c4  = f & 7;
            *(float4*)(out + (size_t)row * NPIX + tb + c4 * 4) = z;
        }
    }

    const int hi   = lane >> 4;                 // 0 -> K{0,1}, 1 -> K{2,3}
    const int nl   = lane & 15;
    const int arow = wave * 16 + nl;            // A row this lane supplies
    const int koff = hi * 2;

    v8f acc0 = {};
    v8f acc1 = {};

    for (int k0 = 0; k0 < NV; k0 += KB) {
        // ---- stage A chunk: 128 rows x 64 k  (8 float4 per thread) ----
        #pragma unroll
        for (int i = 0; i < 8; ++i) {
            int f   = tid + i * 256;
            int row = f >> 4;                   // f / 16
            int c4  = f & 15;
            float4 v = *(const float4*)(X + (size_t)row * NPIX + k0 + c4 * 4);
            *(float4*)(Alds + row * SA + c4 * 4) = v;
        }
        // ---- stage B chunk transposed: global row-major read, Bt[n][k] write ----
        #pragma unroll
        for (int i = 0; i < 2; ++i) {
            int f  = tid + i * 256;
            int k  = f >> 3;                    // f / 8   -> K row
            int c4 = f & 7;                     // 4-col group
            float4 v = *(const float4*)(R + (size_t)(k0 + k) * NV + nbase + c4 * 4);
            Btld[(c4 * 4 + 0) * ST + k] = v.x;
            Btld[(c4 * 4 + 1) * ST + k] = v.y;
            Btld[(c4 * 4 + 2) * ST + k] = v.z;
            Btld[(c4 * 4 + 3) * ST + k] = v.w;
        }
        // prefetch next B chunk (global_prefetch_b8 on gfx1250)
        if (k0 + KB < NV) {
            __builtin_prefetch(R + (size_t)(k0 + KB + (tid >> 3)) * NV + nbase, 0, 1);
        }
        __syncthreads();

        // ---- 16 k-steps of V_WMMA_F32_16X16X4_F32 per chunk, 2 col-tiles ----
        #pragma unroll
        for (int kk = 0; kk < KB; kk += 4) {
            // A 16x4 frag: lanes 0-15 hold K=kk..kk+1, lanes 16-31 K=kk+2..kk+3
            v2f a  = *(const v2f*)(Alds + arow * SA + kk + koff);
            // B 4x16 frags: contiguous b64 from transposed LDS, already in
            // even-aligned pair order (no repack movs)
            v2f b0 = *(const v2f*)(Btld + nl * ST        + kk + koff);
            v2f b1 = *(const v2f*)(Btld + (nl + 16) * ST + kk + koff);
            acc0 = __builtin_amdgcn_wmma_f32_16x16x4_f32(
                       false, a, false, b0, (short)0, acc0, false, false);
            acc1 = __builtin_amdgcn_wmma_f32_16x16x4_f32(
                       false, a, false, b1, (short)0, acc1, false, false);
        }
        __syncthreads();
    }

    // ---- clip to [0,1] and store; C/D layout: VGPR i -> M = i + 8*hi, N = nl ----
    const int orow = wave * 16 + hi * 8;
    #pragma unroll
    for (int i = 0; i < 8; ++i) {
        float v0 = fminf(fmaxf(acc0[i], 0.f), 1.f);
        float v1 = fminf(fmaxf(acc1[i], 0.f), 1.f);
        out[(size_t)(orow + i) * NPIX + nbase + nl]      = v0;
        out[(size_t)(orow + i) * NPIX + nbase + 16 + nl] = v1;
    }
}

extern "C" void kernel_launch(void* const* d_in, const int* in_sizes, int n_in,
                              void* d_out, int out_size, void* d_ws, size_t ws_size,
                              hipStream_t stream) {
    const float* x = (const float*)d_in[0];        // (8,16,12288) fp32
    const float* r = (const float*)d_in[1];        // (6144,6144)  fp32
    float* out     = (float*)d_out;                // (8,16,12288) fp32

    dim3 grid(NV / COLS);   // 192 blocks, each owns a 32-column slice of B
    dim3 block(256);        // 8 wave32 waves
    ellipse_mix_gemm<<<grid, block, 0, stream>>>(x, r, out);
}